// MultiResnet_44152263803452
// MI455X (gfx1250) — compile-verified
//
#include <hip/hip_runtime.h>

typedef __attribute__((ext_vector_type(2))) float v2f;
typedef __attribute__((ext_vector_type(8))) float v8f;

#define NMODELS 64
#define NB      65536
#define FIN     6
#define HID     64
#define FOUT    3
#define CHUNK   128
#define MAXCHUNK 512
#define HS      20   /* padded activation row stride (banks: 2*20%64=40, 8*20%64=32 -> conflict free) */
#define W0S     9    /* padded layer0 weight row stride */
#define W1S     65   /* padded 64x64 weight row stride */

/* d_out layout (floats, concatenated in reference return order) */
#define OUT_MO   0
#define OUT_IDX  (NB*2*FOUT)
#define OUT_LOG  (OUT_IDX + NB)
#define OUT_PROB (OUT_LOG + NB*NMODELS)

__global__ void k_zero(int* __restrict__ p) {
  int t = blockIdx.x * blockDim.x + threadIdx.x;
  if (t < 128) p[t] = 0;   /* counts[64] + cursor[64] */
}

__global__ void k_select(const float* __restrict__ in, float* __restrict__ out,
                         int* __restrict__ counts, int* __restrict__ sidx) {
  int b = blockIdx.x * blockDim.x + threadIdx.x;
  if (b >= NB) return;
  float x = in[b * FIN + 0];
  float z = in[b * FIN + 2];
  const float TWO_PI = 6.28318530717958647692f;
  float ang = atan2f(z, x);
  float a = fmodf(ang + TWO_PI, TWO_PI) * ((float)NMODELS / TWO_PI);
  int idx = (int)floorf(a);
  idx = idx < 0 ? 0 : (idx > NMODELS - 1 ? NMODELS - 1 : idx);
  sidx[b] = idx;
  atomicAdd(&counts[idx], 1);
  out[OUT_IDX + b] = (float)idx;

  /* constant logits / softmax(ones) fills — the bandwidth-dominant part, use b128 stores */
  const float4 ones = make_float4(1.f, 1.f, 1.f, 1.f);
  const float pc = 1.0f / (float)NMODELS;
  const float4 pv = make_float4(pc, pc, pc, pc);
  float4* lg = (float4*)(out + OUT_LOG + (size_t)b * NMODELS);
  float4* pr = (float4*)(out + OUT_PROB + (size_t)b * NMODELS);
#pragma unroll
  for (int i = 0; i < NMODELS / 4; i++) { lg[i] = ones; pr[i] = pv; }
}

__global__ void k_scan(int* __restrict__ ws) {
  if (threadIdx.x == 0 && blockIdx.x == 0) {
    int acc = 0;
    for (int i = 0; i < NMODELS; i++) {
      int c = ws[i];        /* counts  */
      ws[128 + i] = acc;    /* offsets */
      ws[64 + i] = acc;     /* cursor  */
      acc += c;
    }
  }
}

__global__ void k_scatter(int* __restrict__ cursor, const int* __restrict__ sidx,
                          int* __restrict__ order) {
  int b = blockIdx.x * blockDim.x + threadIdx.x;
  if (b >= NB) return;
  int pos = atomicAdd(&cursor[sidx[b]], 1);
  order[pos] = b;
}

__global__ void __launch_bounds__(256) k_mlp(
    const float* __restrict__ in,
    const float* __restrict__ W0, const float* __restrict__ b0,
    const float* __restrict__ W1, const float* __restrict__ b1,
    const float* __restrict__ W2, const float* __restrict__ b2,
    const float* __restrict__ W3, const float* __restrict__ b3,
    const float* __restrict__ Wf, const float* __restrict__ bf,
    const int* __restrict__ counts, const int* __restrict__ offsets,
    const int* __restrict__ order, float* __restrict__ out) {
  __shared__ float sW[HID * W1S];     /* current 64x64 weights, padded rows   */
  __shared__ float sW0[HID * W0S];    /* layer0 weights padded K=6 -> 8 (+1)  */
  __shared__ float sB[4 * HID];
  __shared__ float sWf[FOUT * HID];
  __shared__ float sbf[4];
  __shared__ float sH[8][HID * HS];   /* per-wave activation tiles 64 x 16    */

  const int m = blockIdx.x;
  const int cnt = counts[m];
  const int base = blockIdx.y * CHUNK;
  if (base >= cnt) return;            /* uniform early exit (before barriers) */
  const int off = offsets[m];
  const int tid = threadIdx.x;

  /* one-time small weights */
  for (int i = tid; i < HID * W0S; i += 256) {
    int r = i / W0S, c = i % W0S;
    sW0[i] = (c < FIN) ? W0[(size_t)m * HID * FIN + r * FIN + c] : 0.f;
  }
  if (tid < HID) {
    sB[0 * HID + tid] = b0[m * HID + tid];
    sB[1 * HID + tid] = b1[m * HID + tid];
    sB[2 * HID + tid] = b2[m * HID + tid];
    sB[3 * HID + tid] = b3[m * HID + tid];
  }
  if (tid < FOUT * HID) sWf[tid] = Wf[m * FOUT * HID + tid];
  if (tid < FOUT) sbf[tid] = bf[m * FOUT + tid];

  const int wave = tid >> 5;
  const int lane = tid & 31;
  const int half = lane >> 4;
  const int col = lane & 15;
  const int srel = base + wave * 16;
  const bool active = srel < cnt;     /* wave-uniform -> EXEC all-ones inside */
  int navail = cnt - srel;
  if (navail > 16) navail = 16;
  if (navail < 0) navail = 0;
  float* H = sH[wave];

  /* one 64x16-output layer: D = W(64xK) * X(Kx16), fp32 WMMA, K accumulated 4 at a time */
  auto wmma_layer = [&](const float* w, int wstr, const float* bias, int ksteps) {
    v8f c0 = {0.f, 0.f, 0.f, 0.f, 0.f, 0.f, 0.f, 0.f};
    v8f c1 = c0, c2 = c0, c3 = c0;
    for (int k = 0; k < ksteps; k++) {
      int kb = k * 4;
      v2f bm;  /* B 4x16: v0 = rows {K, K+2} by lane half, v1 = rows {K+1, K+3} */
      bm.x = H[(kb + 2 * half + 0) * HS + col];
      bm.y = H[(kb + 2 * half + 1) * HS + col];
#define MT(CREG, MTI)                                                              \
      { v2f a; int row = (MTI)*16 + col;                                           \
        a.x = w[row * wstr + kb + 2 * half];                                       \
        a.y = w[row * wstr + kb + 2 * half + 1];                                   \
        CREG = __builtin_amdgcn_wmma_f32_16x16x4_f32(false, a, false, bm,          \
                                                     (short)0, CREG, false, false); }
      MT(c0, 0) MT(c1, 1) MT(c2, 2) MT(c3, 3)
#undef MT
    }
    /* bias + ReLU epilogue; D vgpr r holds rows {r, r+8} by lane half */
#pragma unroll
    for (int r = 0; r < 8; r++) {
      int rr = r + 8 * half;
      float v;
      v = c0[r] + bias[0 * 16 + rr]; H[(0 * 16 + rr) * HS + col] = fmaxf(v, 0.f);
      v = c1[r] + bias[1 * 16 + rr]; H[(1 * 16 + rr) * HS + col] = fmaxf(v, 0.f);
      v = c2[r] + bias[2 * 16 + rr]; H[(2 * 16 + rr) * HS + col] = fmaxf(v, 0.f);
      v = c3[r] + bias[3 * 16 + rr]; H[(3 * 16 + rr) * HS + col] = fmaxf(v, 0.f);
    }
  };

  /* 3-wide head, evaluated at t=0 (after layer1) and t=1 (after layer3) */
  auto emit = [&](int t) {
    for (int p = lane; p < 16 * FOUT; p += 32) {
      int s = p / FOUT, o = p - s * FOUT;
      float a = sbf[o];
#pragma unroll
      for (int k = 0; k < HID; k++) a += sWf[o * HID + k] * H[k * HS + s];
      if (s < navail) {
        int sid = order[off + srel + s];
        out[(size_t)sid * (2 * FOUT) + t * FOUT + o] = a;
      }
    }
  };

  __syncthreads();  /* sW0 / sB / sWf ready */

  if (active) {
    /* gather 16 same-model inputs into X (rows=features padded to 8, cols=samples) */
    for (int tix = lane; tix < 8 * 16; tix += 32) {
      int r = tix >> 4, c = tix & 15;
      int sid = order[off + srel + (c < navail ? c : (navail - 1))];
      H[r * HS + c] = (r < FIN) ? in[(size_t)sid * FIN + r] : 0.f;
    }
    wmma_layer(sW0, W0S, sB + 0 * HID, 2);          /* layer 0: 6->64, K padded to 8 */
  }

  __syncthreads();
  for (int i = tid; i < HID * HID; i += 256) { int r = i >> 6, c = i & 63; sW[r * W1S + c] = W1[(size_t)m * HID * HID + i]; }
  __syncthreads();
  if (active) { wmma_layer(sW, W1S, sB + 1 * HID, 16); emit(0); }

  __syncthreads();
  for (int i = tid; i < HID * HID; i += 256) { int r = i >> 6, c = i & 63; sW[r * W1S + c] = W2[(size_t)m * HID * HID + i]; }
  __syncthreads();
  if (active) wmma_layer(sW, W1S, sB + 2 * HID, 16);

  __syncthreads();
  for (int i = tid; i < HID * HID; i += 256) { int r = i >> 6, c = i & 63; sW[r * W1S + c] = W3[(size_t)m * HID * HID + i]; }
  __syncthreads();
  if (active) { wmma_layer(sW, W1S, sB + 3 * HID, 16); emit(1); }
}

extern "C" void kernel_launch(void* const* d_in, const int* in_sizes, int n_in,
                              void* d_out, int out_size, void* d_ws, size_t ws_size,
                              hipStream_t stream) {
  (void)in_sizes; (void)n_in; (void)out_size; (void)ws_size;
  const float* in = (const float*)d_in[0];
  const float* W0 = (const float*)d_in[1];
  const float* b0 = (const float*)d_in[2];
  const float* W1 = (const float*)d_in[3];
  const float* b1 = (const float*)d_in[4];
  const float* W2 = (const float*)d_in[5];
  const float* b2 = (const float*)d_in[6];
  const float* W3 = (const float*)d_in[7];
  const float* b3 = (const float*)d_in[8];
  const float* Wf = (const float*)d_in[9];
  const float* bf = (const float*)d_in[10];
  float* out = (float*)d_out;

  int* wsi = (int*)d_ws;
  int* counts = wsi;            /* [0,64)          */
  int* cursor = wsi + 64;       /* [64,128)        */
  int* offsets = wsi + 128;     /* [128,192)       */
  int* sidx = wsi + 192;        /* [192,192+NB)    */
  int* order = wsi + 192 + NB;  /* [.., +NB)       */

  k_zero<<<1, 128, 0, stream>>>(wsi);
  k_select<<<NB / 256, 256, 0, stream>>>(in, out, counts, sidx);
  k_scan<<<1, 64, 0, stream>>>(wsi);
  k_scatter<<<NB / 256, 256, 0, stream>>>(cursor, sidx, order);
  k_mlp<<<dim3(NMODELS, MAXCHUNK), 256, 0, stream>>>(
      in, W0, b0, W1, b1, W2, b2, W3, b3, Wf, bf, counts, offsets, order, out);
}